// ReweightedMarginLoss_29618094474149
// MI455X (gfx1250) — compile-verified
//
#include <hip/hip_runtime.h>
#include <math.h>

typedef float v2f __attribute__((ext_vector_type(2)));
typedef float v8f __attribute__((ext_vector_type(8)));

#define BDIM 32768
#define CDIM 512
#define COLG 16                                   // columns per wave group
#define ROWCHUNKS 128                             // grid.y
#define WAVES 8                                   // waves per 256-thread block
#define ROWS_PER_BLOCK (BDIM / ROWCHUNKS)         // 256
#define ROWS_PER_WAVE  (ROWS_PER_BLOCK / WAVES)   // 32

__global__ void rml_init_ws(float* ws, int n) {
    int i = blockIdx.x * blockDim.x + threadIdx.x;
    if (i < n) ws[i] = 0.f;
}

__device__ __forceinline__ float softplus_f(float z) {
    // numerically stable: max(z,0) + log(1 + exp(-|z|))
    return fmaxf(z, 0.f) + __logf(1.f + __expf(-fabsf(z)));
}

__global__ void __launch_bounds__(256, 2)
rml_partial(const float* __restrict__ pred, const int* __restrict__ ty,
            const float* __restrict__ cnums, float* __restrict__ ws)
{
    const int wave  = threadIdx.x >> 5;
    const int lane  = threadIdx.x & 31;
    const int nloc  = lane & 15;            // N index within 16-col group
    const int khalf = (lane >> 4) << 1;     // rows K0/K1 (lanes 0-15) or K2/K3 (16-31)
    const int colbase = blockIdx.x * COLG;
    const int col     = colbase + nloc;
    const int row0    = blockIdx.y * ROWS_PER_BLOCK + wave * ROWS_PER_WAVE;

    // delta[c] = c_nums[c,0]^-0.25 = rsqrt(sqrt(x))
    const float delta = rsqrtf(sqrtf(cnums[(size_t)col * 2]));

    v2f a; a.x = 1.f; a.y = 1.f;            // A = ones(16x4): C[m][n] += sum_k B[k][n]
    v8f cp = {}; v8f cn = {}; v8f cP = {}; v8f cN = {};

    const float* p0 = pred + (size_t)(row0 + khalf) * CDIM + col;
    const int*   t0 = ty   + (size_t)(row0 + khalf) * CDIM + col;

    for (int r = 0; r < ROWS_PER_WAVE; r += 4) {
        const float* p = p0 + (size_t)r * CDIM;
        const int*   t = t0 + (size_t)r * CDIM;
        __builtin_prefetch(p + 4 * CDIM, 0, 1);   // global_prefetch_b8 on next tile

        float x0 = p[0], x1 = p[CDIM];
        int   y0 = t[0], y1 = t[CDIM];

        // one softplus per element, routed by label
        float z0 = (y0 == 1) ? (delta - x0) : (x0 + delta);
        float z1 = (y1 == 1) ? (delta - x1) : (x1 + delta);
        float s0 = softplus_f(z0);
        float s1 = softplus_f(z1);

        v2f bp, bn, bP, bN;
        bp.x = (y0 == 1) ? s0  : 0.f;  bp.y = (y1 == 1) ? s1  : 0.f;
        bn.x = (y0 == 1) ? 0.f : s0;   bn.y = (y1 == 1) ? 0.f : s1;
        bP.x = (y0 == 1) ? 1.f : 0.f;  bP.y = (y1 == 1) ? 1.f : 0.f;
        bN.x = 1.f - bP.x;             bN.y = 1.f - bP.y;

        cp = __builtin_amdgcn_wmma_f32_16x16x4_f32(false, a, false, bp, (short)0, cp, false, false);
        cn = __builtin_amdgcn_wmma_f32_16x16x4_f32(false, a, false, bn, (short)0, cn, false, false);
        cP = __builtin_amdgcn_wmma_f32_16x16x4_f32(false, a, false, bP, (short)0, cP, false, false);
        cN = __builtin_amdgcn_wmma_f32_16x16x4_f32(false, a, false, bN, (short)0, cN, false, false);
    }

    // C/D layout: VGPR0 on lanes 0-15 = row M=0, N=lane -> column sums for this wave
    __shared__ float red[WAVES][COLG][4];
    if (lane < 16) {
        red[wave][lane][0] = cp[0];
        red[wave][lane][1] = cn[0];
        red[wave][lane][2] = cP[0];
        red[wave][lane][3] = cN[0];
    }
    __syncthreads();

    if (threadIdx.x < COLG * 4) {
        int c = threadIdx.x >> 2;
        int k = threadIdx.x & 3;
        float s = 0.f;
        #pragma unroll
        for (int w = 0; w < WAVES; ++w) s += red[w][c][k];
        atomicAdd(&ws[(size_t)(colbase + c) * 4 + k], s);
    }
}

__global__ void rml_final(const float* __restrict__ ws, float* __restrict__ out)
{
    const int c = threadIdx.x;   // 512 threads, one per class
    float ps = ws[c * 4 + 0];
    float ns = ws[c * 4 + 1];
    float P  = ws[c * 4 + 2];
    float N  = ws[c * 4 + 3];

    float per   = ps / fmaxf(P, 1.f) + ns / fmaxf(N, 1.f);
    bool  valid = (P > 0.f) && (N > 0.f);
    float contrib = valid ? per : 0.f;
    float cnt     = valid ? 1.f : 0.f;

    __shared__ float ssum[CDIM];
    __shared__ float scnt[CDIM];
    ssum[c] = contrib;
    scnt[c] = cnt;
    __syncthreads();
    for (int off = CDIM / 2; off > 0; off >>= 1) {
        if (c < off) { ssum[c] += ssum[c + off]; scnt[c] += scnt[c + off]; }
        __syncthreads();
    }
    if (c == 0) out[0] = ssum[0] / fmaxf(scnt[0], 1.f);
}

extern "C" void kernel_launch(void* const* d_in, const int* in_sizes, int n_in,
                              void* d_out, int out_size, void* d_ws, size_t ws_size,
                              hipStream_t stream) {
    const float* pred = (const float*)d_in[0];
    const int*   ty   = (const int*)d_in[1];
    const float* cn   = (const float*)d_in[2];
    float* out = (float*)d_out;
    float* ws  = (float*)d_ws;

    (void)in_sizes; (void)n_in; (void)out_size; (void)ws_size;

    rml_init_ws<<<(CDIM * 4 + 255) / 256, 256, 0, stream>>>(ws, CDIM * 4);
    rml_partial<<<dim3(CDIM / COLG, ROWCHUNKS), 256, 0, stream>>>(pred, ty, cn, ws);
    rml_final<<<1, CDIM, 0, stream>>>(ws, out);
}